// GAT_17231408791969
// MI455X (gfx1250) — compile-verified
//
#include <hip/hip_runtime.h>
#include <math.h>

// ---------------------------------------------------------------------------
// GAT on MI455X (gfx1250, wave32).
// GEMM h@W uses native fp32 WMMA: V_WMMA_F32_16X16X4_F32.
// Edge softmax/aggregation via L2-resident atomics (h fits in 192MB L2).
// ---------------------------------------------------------------------------

typedef float v2f __attribute__((ext_vector_type(2)));
typedef float v8f __attribute__((ext_vector_type(8)));

#define HIDF 128   // feature width (all layers 128)
#define OUTF 64
#define NGRAPH 64

// ---- order-preserving float<->int map for atomic segment-max --------------
__device__ __forceinline__ int f2ord(float f) {
    int i = __float_as_int(f);
    return i ^ ((i >> 31) & 0x7fffffff);   // involution, monotone in float order
}
__device__ __forceinline__ float ord2f(int i) {
    return __int_as_float(i ^ ((i >> 31) & 0x7fffffff));
}

// ---- h0 = emb[x] ----------------------------------------------------------
__global__ void gat_gather_emb(const int* __restrict__ x,
                               const float* __restrict__ emb,
                               float* __restrict__ H, int n) {
    int i = blockIdx.x * blockDim.x + threadIdx.x;
    if (i >= n * HIDF) return;
    int node = i >> 7, c = i & (HIDF - 1);
    H[i] = emb[(size_t)x[node] * HIDF + c];
}

// ---- H1 = H0 @ W  (M x 128 @ 128 x 128), fp32 WMMA ------------------------
// One wave -> 16 rows x 128 cols. A: 16x4 f32 (2 VGPR/lane), B: 4x16 (2 VGPR),
// C/D: 16x16 f32 (8 VGPR). 8 N-tiles accumulated over K=128 in steps of 4.
__global__ void gat_gemm_hw(const float* __restrict__ H,
                            const float* __restrict__ W,
                            float* __restrict__ O, int nrows) {
    __shared__ float sW[HIDF * HIDF];   // 64 KB: whole weight matrix per block
    for (int i = threadIdx.x; i < HIDF * HIDF; i += blockDim.x)
        sW[i] = W[i];
    __syncthreads();

    int wave = (blockIdx.x * blockDim.x + threadIdx.x) >> 5;
    int lane = threadIdx.x & 31;
    int row0 = wave * 16;
    if (row0 >= nrows) return;          // wave-uniform: EXEC stays all-1s

    int half = lane >> 4;               // 0: K pair {0,1}, 1: K pair {2,3}
    int l16  = lane & 15;

    v8f acc[8];
    v8f zero = {0.f, 0.f, 0.f, 0.f, 0.f, 0.f, 0.f, 0.f};
#pragma unroll
    for (int t = 0; t < 8; ++t) acc[t] = zero;

    const float* Arow = H + (size_t)(row0 + l16) * HIDF;  // row M = l16
    for (int k = 0; k < HIDF; k += 4) {
        v2f a;
        a.x = Arow[k + 2 * half];       // K = k + 2*half
        a.y = Arow[k + 2 * half + 1];   // K = k + 2*half + 1
#pragma unroll
        for (int t = 0; t < 8; ++t) {
            v2f b;                       // B[K][N], N = t*16 + l16
            b.x = sW[(k + 2 * half) * HIDF + t * 16 + l16];
            b.y = sW[(k + 2 * half + 1) * HIDF + t * 16 + l16];
            acc[t] = __builtin_amdgcn_wmma_f32_16x16x4_f32(
                false, a, false, b, (short)0, acc[t], false, false);
        }
    }
    // D layout: VGPR i -> M = i + 8*half, lane(l16) -> N
    int mbase = row0 + 8 * half;
#pragma unroll
    for (int t = 0; t < 8; ++t)
#pragma unroll
        for (int i = 0; i < 8; ++i)
            O[(size_t)(mbase + i) * HIDF + t * 16 + l16] = acc[t][i];
}

// ---- s_src[n] = h[n].a_src ; s_dst[n] = h[n].a_dst (one wave / node) ------
__global__ void gat_scores(const float* __restrict__ H,
                           const float* __restrict__ asrc,
                           const float* __restrict__ adst,
                           float* __restrict__ ssrc, float* __restrict__ sdst,
                           int n) {
    int wave = (blockIdx.x * blockDim.x + threadIdx.x) >> 5;
    int lane = threadIdx.x & 31;
    if (wave >= n) return;
    const float* h = H + (size_t)wave * HIDF;
    float s0 = 0.f, s1 = 0.f;
#pragma unroll
    for (int c = 0; c < HIDF; c += 32) {
        float v = h[c + lane];
        s0 += v * asrc[c + lane];
        s1 += v * adst[c + lane];
    }
#pragma unroll
    for (int off = 16; off > 0; off >>= 1) {
        s0 += __shfl_down(s0, off, 32);
        s1 += __shfl_down(s1, off, 32);
    }
    if (lane == 0) { ssrc[wave] = s0; sdst[wave] = s1; }
}

// ---- per-node init for softmax --------------------------------------------
__global__ void gat_init_nodes(int* __restrict__ maxb, float* __restrict__ denom, int n) {
    int i = blockIdx.x * blockDim.x + threadIdx.x;
    if (i >= n) return;
    maxb[i] = f2ord(-INFINITY);
    denom[i] = 0.f;
}

__global__ void gat_zero_f(float* __restrict__ p, long long n) {
    long long i = (long long)blockIdx.x * blockDim.x + threadIdx.x;
    if (i < n) p[i] = 0.f;
}

// ---- e = leaky_relu(s_src[src]+s_dst[dst]); segment max -------------------
__global__ void gat_edge_logit_max(const int* __restrict__ src, const int* __restrict__ dst,
                                   const float* __restrict__ ssrc, const float* __restrict__ sdst,
                                   float* __restrict__ elog, int* __restrict__ maxb, int E) {
    int e = blockIdx.x * blockDim.x + threadIdx.x;
    if (e >= E) return;
    int d = dst[e];
    float v = ssrc[src[e]] + sdst[d];
    v = (v >= 0.f) ? v : 0.2f * v;
    elog[e] = v;
    atomicMax(&maxb[d], f2ord(v));
}

// ---- w = exp(e - m[dst]); denom = segment sum -----------------------------
__global__ void gat_edge_exp(const int* __restrict__ dst,
                             float* __restrict__ ew,
                             const int* __restrict__ maxb,
                             float* __restrict__ denom, int E) {
    int e = blockIdx.x * blockDim.x + threadIdx.x;
    if (e >= E) return;
    int d = dst[e];
    float w = expf(ew[e] - ord2f(maxb[d]));
    ew[e] = w;
    atomicAdd(&denom[d], w);
}

// ---- out[dst] += (w/denom[dst]) * h[src]  (one wave / edge, float4/lane) --
__global__ void gat_edge_aggregate(const int* __restrict__ src, const int* __restrict__ dst,
                                   const float* __restrict__ ew, const float* __restrict__ denom,
                                   const float* __restrict__ H, float* __restrict__ O, int E) {
    int wave = (blockIdx.x * blockDim.x + threadIdx.x) >> 5;
    int lane = threadIdx.x & 31;
    if (wave >= E) return;
    int s = src[wave], d = dst[wave];
    float coef = ew[wave] / denom[d];
    const float4* hs = (const float4*)(H + (size_t)s * HIDF);
    float4 v = hs[lane];
    float* o = O + (size_t)d * HIDF + lane * 4;
    atomicAdd(o + 0, coef * v.x);
    atomicAdd(o + 1, coef * v.y);
    atomicAdd(o + 2, coef * v.z);
    atomicAdd(o + 3, coef * v.w);
}

// ---- h += bias (per channel); optional relu -------------------------------
__global__ void gat_bias_act(float* __restrict__ H, const float* __restrict__ b,
                             int total, int relu) {
    int i = blockIdx.x * blockDim.x + threadIdx.x;
    if (i >= total) return;
    float v = H[i] + b[i & (HIDF - 1)];
    H[i] = relu ? fmaxf(v, 0.f) : v;
}

// ---- mean pool over graphs ------------------------------------------------
__global__ void gat_pool(const float* __restrict__ H, const int* __restrict__ batch,
                         float* __restrict__ pooled, float* __restrict__ cnt, int n) {
    int wave = (blockIdx.x * blockDim.x + threadIdx.x) >> 5;
    int lane = threadIdx.x & 31;
    if (wave >= n) return;
    int g = batch[wave];
    const float4* h = (const float4*)(H + (size_t)wave * HIDF);
    float4 v = h[lane];
    float* p = pooled + (size_t)g * HIDF + lane * 4;
    atomicAdd(p + 0, v.x);
    atomicAdd(p + 1, v.y);
    atomicAdd(p + 2, v.z);
    atomicAdd(p + 3, v.w);
    if (lane == 0) atomicAdd(&cnt[g], 1.0f);
}

// ---- out = (pooled/cnt) @ Wout + bout  (64x128 @ 128x64, tiny) ------------
__global__ void gat_final(const float* __restrict__ pooled, const float* __restrict__ cnt,
                          const float* __restrict__ Wout, const float* __restrict__ bout,
                          float* __restrict__ out) {
    int idx = blockIdx.x * blockDim.x + threadIdx.x;
    if (idx >= NGRAPH * OUTF) return;
    int g = idx >> 6, o = idx & (OUTF - 1);
    float c = fmaxf(cnt[g], 1.0f);
    float s = 0.f;
    for (int k = 0; k < HIDF; ++k)
        s += (pooled[(size_t)g * HIDF + k] / c) * Wout[k * OUTF + o];
    out[idx] = s + bout[o];
}

// ---------------------------------------------------------------------------
extern "C" void kernel_launch(void* const* d_in, const int* in_sizes, int n_in,
                              void* d_out, int out_size, void* d_ws, size_t ws_size,
                              hipStream_t stream) {
    const int*   x     = (const int*)d_in[0];
    const int*   edges = (const int*)d_in[1];
    const int*   batch = (const int*)d_in[2];
    const float* emb   = (const float*)d_in[3];
    const float* Ws    = (const float*)d_in[4];
    const float* asrc  = (const float*)d_in[5];
    const float* adst  = (const float*)d_in[6];
    const float* bs    = (const float*)d_in[7];
    const float* Wout  = (const float*)d_in[8];
    const float* bout  = (const float*)d_in[9];
    float* out = (float*)d_out;

    const int n = in_sizes[0];          // 50000 nodes (multiple of 16)
    const int E = in_sizes[1] / 2;      // 550000 edges
    const int* src = edges;
    const int* dst = edges + E;

    // workspace carve-up (all 16B aligned)
    char* ws = (char*)d_ws;
    float* hA    = (float*)ws; ws += (size_t)n * HIDF * sizeof(float); // current h
    float* hB    = (float*)ws; ws += (size_t)n * HIDF * sizeof(float); // gemm scratch
    float* ew    = (float*)ws; ws += (size_t)E * sizeof(float);        // edge logits/weights
    float* ssrc  = (float*)ws; ws += (size_t)n * sizeof(float);
    float* sdst  = (float*)ws; ws += (size_t)n * sizeof(float);
    float* denom = (float*)ws; ws += (size_t)n * sizeof(float);
    int*   maxb  = (int*)ws;   ws += (size_t)n * sizeof(int);
    float* pooled= (float*)ws; ws += (size_t)NGRAPH * HIDF * sizeof(float);
    float* cnt   = (float*)ws; ws += (size_t)NGRAPH * sizeof(float);

    const int TPB = 256;
    int gElem  = (n * HIDF + TPB - 1) / TPB;          // node*feature grid
    int gNode  = (n + TPB - 1) / TPB;
    int gNodeW = (n + 7) / 8;                          // 1 wave / node
    int gEdge  = (E + TPB - 1) / TPB;
    int gEdgeW = (E + 7) / 8;                          // 1 wave / edge
    int tiles  = (n + 15) / 16;
    int gGemm  = (tiles + 7) / 8;                      // 8 waves / block

    gat_gather_emb<<<gElem, TPB, 0, stream>>>(x, emb, hA, n);

    for (int l = 0; l < 3; ++l) {
        // hB = hA @ W[l]   (fp32 WMMA)
        gat_gemm_hw<<<gGemm, TPB, 0, stream>>>(hA, Ws + (size_t)l * HIDF * HIDF, hB, n);
        // attention scalars
        gat_scores<<<gNodeW, TPB, 0, stream>>>(hB, asrc + l * HIDF, adst + l * HIDF,
                                               ssrc, sdst, n);
        // edge softmax
        gat_init_nodes<<<gNode, TPB, 0, stream>>>(maxb, denom, n);
        gat_edge_logit_max<<<gEdge, TPB, 0, stream>>>(src, dst, ssrc, sdst, ew, maxb, E);
        gat_edge_exp<<<gEdge, TPB, 0, stream>>>(dst, ew, maxb, denom, E);
        // aggregate into hA (free after gemm): hA = segsum(coef * hB[src])
        gat_zero_f<<<gElem, TPB, 0, stream>>>(hA, (long long)n * HIDF);
        gat_edge_aggregate<<<gEdgeW, TPB, 0, stream>>>(src, dst, ew, denom, hB, hA, E);
        // + bias, relu on hidden layers
        gat_bias_act<<<gElem, TPB, 0, stream>>>(hA, bs + l * HIDF, n * HIDF, l < 2 ? 1 : 0);
    }

    // global mean pool + output projection
    gat_zero_f<<<(NGRAPH * HIDF + TPB - 1) / TPB, TPB, 0, stream>>>(pooled, NGRAPH * HIDF);
    gat_zero_f<<<1, TPB, 0, stream>>>(cnt, NGRAPH);
    gat_pool<<<gNodeW, TPB, 0, stream>>>(hA, batch, pooled, cnt, n);
    gat_final<<<(NGRAPH * OUTF + TPB - 1) / TPB, TPB, 0, stream>>>(pooled, cnt, Wout, bout, out);
}